// EMM_67259187855477
// MI455X (gfx1250) — compile-verified
//
#include <hip/hip_runtime.h>
#include <hip/hip_bf16.h>

// ---------------------------------------------------------------------------
// Problem constants (from the reference)
// ---------------------------------------------------------------------------
#define BB   512
#define HH   2048
#define NN   65536
#define MM   128
#define NBANKS 4
#define NM   ((size_t)NN * MM)
#define BN_EPS 1e-5f

// GEMM tiling
#define KSTEP     32          // K-extent staged into LDS per double-buffer slot
#define AS_STRIDE (KSTEP + 4) // padded A row stride (floats) -> conflict-free ds reads
#define BS_STRIDE (MM + 16)   // padded B row stride (floats) -> halves hit disjoint banks

typedef __attribute__((ext_vector_type(2))) float v2f;
typedef __attribute__((ext_vector_type(8))) float v8f;
typedef __attribute__((ext_vector_type(2))) int   v2i;
typedef __attribute__((ext_vector_type(4))) int   v4i;

// ---------------------------------------------------------------------------
// Async global->LDS copy helpers (gfx1250 GLOBAL_LOAD_ASYNC_TO_LDS_*).
// Builtin signature (from the compiler's own diagnostic):
//   (vNi addrspace(1)* gsrc, vNi addrspace(3)* ldsdst, imm offset, imm cpol)
// ---------------------------------------------------------------------------
#if __has_builtin(__builtin_amdgcn_global_load_async_to_lds_b128) && \
    __has_builtin(__builtin_amdgcn_global_load_async_to_lds_b64) &&  \
    __has_builtin(__builtin_amdgcn_s_wait_asynccnt)
#define HAVE_ASYNC_LDS 1
#else
#define HAVE_ASYNC_LDS 0
#endif

__device__ __forceinline__ void lds_copy16(float* lds_dst, const float* g_src) {
#if HAVE_ASYNC_LDS
    __builtin_amdgcn_global_load_async_to_lds_b128(
        (__attribute__((address_space(1))) v4i*)(g_src),
        (__attribute__((address_space(3))) v4i*)(lds_dst), 0, 0);
#else
    *reinterpret_cast<float4*>(lds_dst) = *reinterpret_cast<const float4*>(g_src);
#endif
}

__device__ __forceinline__ void lds_copy8(float* lds_dst, const float* g_src) {
#if HAVE_ASYNC_LDS
    __builtin_amdgcn_global_load_async_to_lds_b64(
        (__attribute__((address_space(1))) v2i*)(g_src),
        (__attribute__((address_space(3))) v2i*)(lds_dst), 0, 0);
#else
    *reinterpret_cast<float2*>(lds_dst) = *reinterpret_cast<const float2*>(g_src);
#endif
}

// ---------------------------------------------------------------------------
// Workspace layout (floats)
// ---------------------------------------------------------------------------
#define OFF_E      0          // e_t       [512*128]
#define OFF_A      65536      // a_t       [512*128]
#define OFF_WNB    131072     // wnew_b    [512]
#define OFF_ME     131584     // me        [128]
#define OFF_MA     131712     // ma        [128]
#define OFF_SCALE  131840     // mem_bn scale  [128]
#define OFF_SHIFT  131968     // mem_bn shift  [128]
#define OFF_PSUM   132096     // per-block col sums   [512*128]
#define OFF_PSUM2  197632     // per-block col sumsq  [512*128]

// ---------------------------------------------------------------------------
// Split-K WMMA f32 GEMM stage 1: partial sums.
//   Cpart[split][r, c] = sum_{k in K-chunk} A[r,k] * Bm[k,c]
// Block: 16 rows x 128 cols (8 waves, one 16x16 V_WMMA_F32_16X16X4_F32 tile
// per wave). A (16xKSTEP) and B (KSTEPx128) tiles double-buffered in LDS via
// async global->LDS loads; 5 async ops per wave per stage (uniform), so the
// double-buffer wait is s_wait_asynccnt 5.
// ---------------------------------------------------------------------------
__global__ __launch_bounds__(256) void emm_gemm_wmma_f32(
    const float* __restrict__ A, int lda,
    const float* __restrict__ Bbase, int ldb,
    const int* __restrict__ bank_sel, long bank_stride,
    float* __restrict__ Cpart, int K)
{
    const float* __restrict__ Bm =
        bank_sel ? (Bbase + (size_t)(*bank_sel) * (size_t)bank_stride) : Bbase;

    __shared__ float As[2][16 * AS_STRIDE];
    __shared__ float Bs[2][KSTEP * BS_STRIDE];

    const int t     = threadIdx.x;
    const int lane  = t & 31;
    const int wave  = t >> 5;            // 0..7 -> 16-column tile
    const int l16   = lane & 15;
    const int lhalf = lane >> 4;         // 0 or 1

    const int r0 = blockIdx.x * 16;
    const int c0 = wave * 16;

    const int KC     = K / gridDim.y;    // K-chunk of this split
    const int kbase  = blockIdx.y * KC;
    const int nstage = KC / KSTEP;

    // Copy assignments (all 256 threads participate in both A and B):
    //   A tile 16 x 32 floats: one b64 per thread  (row = t>>4, 2 floats)
    //   B tile 32 x 128 floats: four b128 per thread
    const int a_row = t >> 4;
    const int a_c   = (t & 15) * 2;
    const int b_c4  = (t & 31) * 4;

    v8f acc = {};

    auto issue = [&](int s, int buf) {
        const int kc = kbase + s * KSTEP;
        lds_copy8(&As[buf][a_row * AS_STRIDE + a_c],
                  A + (size_t)(r0 + a_row) * (size_t)lda + kc + a_c);
        #pragma unroll
        for (int i = 0; i < 4; ++i) {
            const int br = (t >> 5) + i * 8;              // rows 0..31
            lds_copy16(&Bs[buf][br * BS_STRIDE + b_c4],
                       Bm + (size_t)(kc + br) * (size_t)ldb + b_c4);
        }
        // speculative prefetch two stages ahead (global_prefetch_b8)
        __builtin_prefetch(A + (size_t)(r0 + a_row) * (size_t)lda
                             + kc + 2 * KSTEP + a_c, 0, 0);
        __builtin_prefetch(Bm + (size_t)(kc + 2 * KSTEP + (t >> 5)) * (size_t)ldb
                             + b_c4, 0, 0);
    };

    issue(0, 0);
    for (int s = 0; s < nstage; ++s) {
        const int buf = s & 1;
        if (s + 1 < nstage) {
            issue(s + 1, buf ^ 1);
#if HAVE_ASYNC_LDS
            __builtin_amdgcn_s_wait_asynccnt(5);   // stage s's 5 ops complete
#endif
        } else {
#if HAVE_ASYNC_LDS
            __builtin_amdgcn_s_wait_asynccnt(0);
#endif
        }
        __syncthreads();                            // stage-s tiles visible to all waves

        const float* __restrict__ Ab = &As[buf][l16 * AS_STRIDE + 2 * lhalf];
        const float* __restrict__ Bb = &Bs[buf][(2 * lhalf) * BS_STRIDE + c0 + l16];
        #pragma unroll
        for (int kk = 0; kk < KSTEP; kk += 4) {
            v2f av = *reinterpret_cast<const v2f*>(Ab + kk);   // A[l16, kk+2h .. +1]
            v2f bv;
            bv.x = Bb[(kk    ) * BS_STRIDE];                   // B[kk+2h  , c0+l16]
            bv.y = Bb[(kk + 1) * BS_STRIDE];                   // B[kk+2h+1, c0+l16]
            acc = __builtin_amdgcn_wmma_f32_16x16x4_f32(
                      false, av, false, bv, (short)0, acc, false, false);
        }
        __syncthreads();                            // done with buf before it is refilled
    }

    // partial-sum store: C/D layout VGPR v -> rows v (half 0) / v+8 (half 1)
    const size_t partbase =
        (size_t)blockIdx.y * ((size_t)gridDim.x * 16 * MM);
    #pragma unroll
    for (int v = 0; v < 8; ++v) {
        const int row = r0 + v + 8 * lhalf;
        Cpart[partbase + (size_t)row * MM + c0 + l16] = acc[v];
    }
}

// ---------------------------------------------------------------------------
// Split-K stage 2: deterministic reduction + bias + optional clip(.,0,1)
// ---------------------------------------------------------------------------
__global__ __launch_bounds__(256) void emm_gemm_reduce(
    const float* __restrict__ part, int nsplit,
    const float* __restrict__ bias,
    float* __restrict__ out, int total, int act_clip01)
{
    const int idx = blockIdx.x * blockDim.x + threadIdx.x;
    if (idx >= total) return;
    float s = 0.0f;
    for (int i = 0; i < nsplit; ++i)
        s += part[(size_t)i * total + idx];
    if (bias) s += bias[idx & (MM - 1)];
    if (act_clip01) s = fminf(fmaxf(s, 0.0f), 1.0f);
    out[idx] = s;
}

// ---------------------------------------------------------------------------
// gamma_t[b] = 1 + relu(h_t[b,:] . gamma_w + gamma_b)
// wnew_b[b]  = p / (p * N),  p = (1e-6)^gamma   (exactly the reference math)
// ---------------------------------------------------------------------------
__global__ __launch_bounds__(256) void emm_gamma_kernel(
    const float* __restrict__ h, const float* __restrict__ gw,
    const float* __restrict__ gb, float* __restrict__ wnew_b)
{
    const int b    = blockIdx.x * 8 + (threadIdx.x >> 5);
    const int lane = threadIdx.x & 31;
    float s = 0.0f;
    for (int k = lane; k < HH; k += 32)
        s += h[(size_t)b * HH + k] * gw[k];
    #pragma unroll
    for (int off = 16; off > 0; off >>= 1)
        s += __shfl_down(s, off, 32);
    if (lane == 0) {
        const float gamma = 1.0f + fmaxf(s + gb[0], 0.0f);
        const float p = powf(1e-6f, gamma);
        wnew_b[b] = p / (p * (float)NN);
    }
}

// ---------------------------------------------------------------------------
// me[m] = sum_b wnew_b[b]*e_t[b,m] ; ma[m] likewise (w_new is row-constant)
// ---------------------------------------------------------------------------
__global__ __launch_bounds__(128) void emm_colmix_kernel(
    const float* __restrict__ e, const float* __restrict__ a,
    const float* __restrict__ wnew_b,
    float* __restrict__ me, float* __restrict__ ma)
{
    const int m = threadIdx.x;
    float se = 0.0f, sa = 0.0f;
    for (int b = 0; b < BB; ++b) {
        const float w = wnew_b[b];
        se += w * e[(size_t)b * MM + m];
        sa += w * a[(size_t)b * MM + m];
    }
    me[m] = se;
    ma[m] = sa;
}

// ---------------------------------------------------------------------------
// w_new output: out[b,n] = wnew_b[b]  (float4 stores, pure bandwidth)
// ---------------------------------------------------------------------------
__global__ __launch_bounds__(256) void emm_fill_wnew(
    const float* __restrict__ wnew_b, float* __restrict__ out)
{
    const size_t idx = ((size_t)blockIdx.x * blockDim.x + threadIdx.x) * 4;
    const int b = (int)(idx >> 16);            // N = 65536 per row
    const float v = wnew_b[b];
    float4 f; f.x = v; f.y = v; f.z = v; f.w = v;
    *reinterpret_cast<float4*>(out + idx) = f;
}

// ---------------------------------------------------------------------------
// m_tp1 = clip(m_t*(1-me)+ma, 0, 1) -> staged into output bank slot
// + deterministic per-block column partial sums/sumsq for mem_bn stats
// ---------------------------------------------------------------------------
__global__ __launch_bounds__(256) void emm_memupdate_kernel(
    const float* __restrict__ mem_in, const int* __restrict__ bank_sel,
    const float* __restrict__ me, const float* __restrict__ ma,
    float* __restrict__ out_mem,
    float* __restrict__ psum, float* __restrict__ psum2)
{
    const int bank = *bank_sel;
    const float* __restrict__ mt = mem_in + (size_t)bank * NM;
    float* __restrict__ dst      = out_mem + (size_t)bank * NM;

    const int m   = threadIdx.x & 127;
    const int sub = threadIdx.x >> 7;
    const int r0  = blockIdx.x * 128;
    const float ef = me[m], af = ma[m];

    float s = 0.0f, s2 = 0.0f;
    for (int r = r0 + sub; r < r0 + 128; r += 2) {
        const size_t off = (size_t)r * MM + m;
        float v = mt[off] * (1.0f - ef) + af;
        v = fminf(fmaxf(v, 0.0f), 1.0f);
        dst[off] = v;
        s += v; s2 += v * v;
    }
    __shared__ float ls[256], ls2[256];
    ls[threadIdx.x] = s; ls2[threadIdx.x] = s2;
    __syncthreads();
    if (sub == 0) {
        psum [(size_t)blockIdx.x * MM + m] = ls [m] + ls [m + 128];
        psum2[(size_t)blockIdx.x * MM + m] = ls2[m] + ls2[m + 128];
    }
}

// ---------------------------------------------------------------------------
// mem_bn affine params per column (deterministic sum over the 512 partials)
// ---------------------------------------------------------------------------
__global__ __launch_bounds__(128) void emm_bnparams_kernel(
    const float* __restrict__ psum, const float* __restrict__ psum2,
    const float* __restrict__ bw, const float* __restrict__ bb,
    float* __restrict__ scale, float* __restrict__ shift)
{
    const int m = threadIdx.x;
    float s = 0.0f, s2 = 0.0f;
    for (int i = 0; i < 512; ++i) {
        s  += psum [(size_t)i * MM + m];
        s2 += psum2[(size_t)i * MM + m];
    }
    const float mu  = s  * (1.0f / (float)NN);
    const float var = s2 * (1.0f / (float)NN) - mu * mu;
    const float rs  = rsqrtf(var + BN_EPS);
    const float sc  = bw[m] * rs;
    scale[m] = sc;
    shift[m] = bb[m] - mu * sc;
}

// ---------------------------------------------------------------------------
// bank_bn: per-slot n, stats over (4 banks x 128 cols) = 512 values.
// ---------------------------------------------------------------------------
__global__ __launch_bounds__(128) void emm_bankbn_kernel(
    const float* __restrict__ mem_in, const int* __restrict__ bank_sel,
    const float* __restrict__ scale, const float* __restrict__ shift,
    const float* __restrict__ bnw, const float* __restrict__ bnb,
    float* __restrict__ out_mem)
{
    const int n = blockIdx.x;
    const int m = threadIdx.x;
    const int bank = *bank_sel;

    float v[NBANKS];
    float s = 0.0f, s2 = 0.0f;
    #pragma unroll
    for (int bk = 0; bk < NBANKS; ++bk) {
        const size_t off = (size_t)bk * NM + (size_t)n * MM + m;
        float x;
        if (bk == bank) {
            const float t = out_mem[off];                       // m_tp1 staged here
            x = fminf(fmaxf(scale[m] * t + shift[m], 0.0f), 1.0f);
        } else {
            x = mem_in[off];
        }
        v[bk] = x; s += x; s2 += x * x;
    }

    __shared__ float ls[128], ls2[128];
    ls[m] = s; ls2[m] = s2;
    __syncthreads();
    #pragma unroll
    for (int off = 64; off > 0; off >>= 1) {
        if (m < off) { ls[m] += ls[m + off]; ls2[m] += ls2[m + off]; }
        __syncthreads();
    }
    const float mu  = ls [0] * (1.0f / 512.0f);
    const float var = ls2[0] * (1.0f / 512.0f) - mu * mu;
    const float rs  = rsqrtf(var + BN_EPS);
    const float w = bnw[n], b = bnb[n];
    #pragma unroll
    for (int bk = 0; bk < NBANKS; ++bk) {
        const float y = (v[bk] - mu) * rs * w + b;
        out_mem[(size_t)bk * NM + (size_t)n * MM + m] =
            fminf(fmaxf(y, 0.0f), 1.0f);
    }
}

// ---------------------------------------------------------------------------
// Host-side launcher
// ---------------------------------------------------------------------------
extern "C" void kernel_launch(void* const* d_in, const int* in_sizes, int n_in,
                              void* d_out, int out_size, void* d_ws, size_t ws_size,
                              hipStream_t stream) {
    (void)in_sizes; (void)n_in; (void)out_size; (void)ws_size;

    const float* h_t     = (const float*)d_in[0];
    const int*   bank_no = (const int*)  d_in[1];
    const float* memory  = (const float*)d_in[2];
    const float* w0      = (const float*)d_in[3];
    const float* gamma_w = (const float*)d_in[12];
    const float* gamma_b = (const float*)d_in[13];
    const float* erase_w = (const float*)d_in[14];
    const float* erase_b = (const float*)d_in[15];
    const float* add_w   = (const float*)d_in[16];
    const float* add_b   = (const float*)d_in[17];
    const float* mem_bn_w = (const float*)d_in[18];
    const float* mem_bn_b = (const float*)d_in[19];
    const float* bank_bn_w = (const float*)d_in[20];
    const float* bank_bn_b = (const float*)d_in[21];

    float* out   = (float*)d_out;
    float* out_r = out;                         // [512,128]
    float* out_w = out + (size_t)BB * MM;       // [512,65536]
    float* out_m = out_w + (size_t)BB * NN;     // [4,65536,128]

    float* ws = (float*)d_ws;
    float* ws_e     = ws + OFF_E;
    float* ws_a     = ws + OFF_A;
    float* ws_wnb   = ws + OFF_WNB;
    float* ws_me    = ws + OFF_ME;
    float* ws_ma    = ws + OFF_MA;
    float* ws_scale = ws + OFF_SCALE;
    float* ws_shift = ws + OFF_SHIFT;
    float* ws_psum  = ws + OFF_PSUM;
    float* ws_psum2 = ws + OFF_PSUM2;

    // GEMM partial sums live in the w_new output region (written later).
    float* gpart = out_w;
    const int total = BB * MM;                  // 65536 outputs per GEMM

    // r_t = w0 @ memory[bank_no]   (split-K 32 -> 1024 blocks, HBM-rate stream)
    emm_gemm_wmma_f32<<<dim3(BB / 16, 32), 256, 0, stream>>>(
        w0, NN, memory, MM, bank_no, (long)NM, gpart, NN);
    emm_gemm_reduce<<<(total + 255) / 256, 256, 0, stream>>>(
        gpart, 32, nullptr, out_r, total, /*act=*/0);

    // e_t / a_t projections (clip(relu(x),0,1) == clip(x,0,1))
    emm_gemm_wmma_f32<<<dim3(BB / 16, 4), 256, 0, stream>>>(
        h_t, HH, erase_w, MM, nullptr, 0, gpart, HH);
    emm_gemm_reduce<<<(total + 255) / 256, 256, 0, stream>>>(
        gpart, 4, erase_b, ws_e, total, /*act=*/1);

    emm_gemm_wmma_f32<<<dim3(BB / 16, 4), 256, 0, stream>>>(
        h_t, HH, add_w, MM, nullptr, 0, gpart, HH);
    emm_gemm_reduce<<<(total + 255) / 256, 256, 0, stream>>>(
        gpart, 4, add_b, ws_a, total, /*act=*/1);

    // gamma -> uniform w_new row value
    emm_gamma_kernel<<<BB / 8, 256, 0, stream>>>(h_t, gamma_w, gamma_b, ws_wnb);

    // me/ma (rank-collapsed w_new^T @ e_t / a_t)
    emm_colmix_kernel<<<1, 128, 0, stream>>>(ws_e, ws_a, ws_wnb, ws_me, ws_ma);

    // w_new output fill (134 MB, bandwidth) -- overwrites the partial scratch
    emm_fill_wnew<<<(BB * (NN / 4)) / 256, 256, 0, stream>>>(ws_wnb, out_w);

    // m_tp1 + mem_bn partial stats
    emm_memupdate_kernel<<<NN / 128, 256, 0, stream>>>(
        memory, bank_no, ws_me, ws_ma, out_m, ws_psum, ws_psum2);

    // mem_bn affine params
    emm_bnparams_kernel<<<1, 128, 0, stream>>>(
        ws_psum, ws_psum2, mem_bn_w, mem_bn_b, ws_scale, ws_shift);

    // bank_bn + final clipped memory output
    emm_bankbn_kernel<<<NN, 128, 0, stream>>>(
        memory, bank_no, ws_scale, ws_shift, bank_bn_w, bank_bn_b, out_m);
}